// SystemGNN_81192061764390
// MI455X (gfx1250) — compile-verified
//
#include <hip/hip_runtime.h>
#include <math.h>

#define N_NODES 20000
#define E_EDGES 320000
#define E_TOT   340000      // E_EDGES + N_NODES self loops
#define D       256
#define NEG_SLOPE 0.2f
#define MT      5           // M tiles per wave in GEMM (1250 % 5 == 0)

typedef __attribute__((ext_vector_type(2))) float v2f;
typedef __attribute__((ext_vector_type(8))) float v8f;

// ---- ordered-uint encoding of float for atomic max (monotonic) ----
__device__ __forceinline__ unsigned f2ord(float f) {
  unsigned u = __float_as_uint(f);
  return (u & 0x80000000u) ? ~u : (u | 0x80000000u);
}
__device__ __forceinline__ float ord2f(unsigned u) {
  return (u & 0x80000000u) ? __uint_as_float(u & 0x7FFFFFFFu)
                           : __uint_as_float(~u);
}

// ---- 1) row-normalize x (one wave32 per node) ----
__global__ void k_normalize(const float* __restrict__ x, float* __restrict__ xn) {
  int wid  = (blockIdx.x * blockDim.x + threadIdx.x) >> 5;
  int lane = threadIdx.x & 31;
  if (wid >= N_NODES) return;
  const float* row = x + (size_t)wid * D;
  float ss = 0.f;
#pragma unroll
  for (int j = 0; j < D / 32; ++j) { float v = row[lane + 32 * j]; ss += v * v; }
#pragma unroll
  for (int off = 16; off; off >>= 1) ss += __shfl_xor(ss, off, 32);
  float nrm = sqrtf(ss);
  if (nrm == 0.f) nrm = 1e-8f;
  float inv = 1.0f / nrm;
  float* orow = xn + (size_t)wid * D;
#pragma unroll
  for (int j = 0; j < D / 32; ++j) orow[lane + 32 * j] = row[lane + 32 * j] * inv;
}

// ---- 2) h = A(N x 256) * W(256 x 256), fp32 WMMA 16x16x4 ----
// block (32,4): 4 waves; each wave computes MT stacked 16x16 C tiles for one
// N-tile, so each B fragment load feeds MT WMMAs.
__global__ void k_gemm(const float* __restrict__ A, const float* __restrict__ W,
                       float* __restrict__ C) {
  int tm0  = blockIdx.x * MT;                   // first M tile (grid.x = 250)
  int tn   = blockIdx.y * 4 + threadIdx.y;      // 0..15 (N tile)
  int lane = threadIdx.x;                       // 0..31
  int half = lane >> 4;                         // K-pair selector
  int l16  = lane & 15;
  int koff = half * 2;

  const float* bcol = W + (tn * 16 + l16);      // B column base
  const float* arow[MT];
#pragma unroll
  for (int i = 0; i < MT; ++i)
    arow[i] = A + (size_t)((tm0 + i) * 16 + l16) * D;

  v8f c[MT] = {};
  for (int k0 = 0; k0 < D; k0 += 4) {
    // B frag: lane l holds W[k0+koff, N=l16], W[k0+koff+1, N=l16]
    v2f b;
    b.x = bcol[(size_t)(k0 + koff) * D];
    b.y = bcol[(size_t)(k0 + koff + 1) * D];
#pragma unroll
    for (int i = 0; i < MT; ++i) {
      // A frag: lane l holds A[M=l16, k0+koff], A[M=l16, k0+koff+1]
      v2f a = *(const v2f*)(arow[i] + k0 + koff);
      c[i] = __builtin_amdgcn_wmma_f32_16x16x4_f32(false, a, false, b,
                                                   (short)0, c[i], false, false);
    }
  }
  // C/D layout: vgpr v -> M = v + (half?8:0), N = l16
#pragma unroll
  for (int i = 0; i < MT; ++i) {
    int rowbase = (tm0 + i) * 16 + half * 8;
    float* crow = C + (size_t)rowbase * D + tn * 16 + l16;
#pragma unroll
    for (int v = 0; v < 8; ++v) crow[(size_t)v * D] = c[i][v];
  }
}

// ---- 3) s[i] = h[i]·a_src, d[i] = h[i]·a_dst (one wave per node) ----
__global__ void k_sd(const float* __restrict__ h, const float* __restrict__ a_s,
                     const float* __restrict__ a_d, float* __restrict__ s,
                     float* __restrict__ d) {
  int wid  = (blockIdx.x * blockDim.x + threadIdx.x) >> 5;
  int lane = threadIdx.x & 31;
  if (wid >= N_NODES) return;
  const float* row = h + (size_t)wid * D;
  float ss = 0.f, dd = 0.f;
#pragma unroll
  for (int j = 0; j < D / 32; ++j) {
    float v = row[lane + 32 * j];
    ss += v * a_s[lane + 32 * j];
    dd += v * a_d[lane + 32 * j];
  }
#pragma unroll
  for (int off = 16; off; off >>= 1) {
    ss += __shfl_xor(ss, off, 32);
    dd += __shfl_xor(dd, off, 32);
  }
  if (lane == 0) { s[wid] = ss; d[wid] = dd; }
}

// ---- per-layer node scratch init ----
__global__ void k_init_nodes(unsigned* __restrict__ m, float* __restrict__ denom) {
  int i = blockIdx.x * blockDim.x + threadIdx.x;
  if (i < N_NODES) { m[i] = 0x007FFFFFu /* f2ord(-inf) */; denom[i] = 0.f; }
}

// init scatter accumulator with broadcast bias (folds "+ b" into the sum)
__global__ void k_init_bias(float* __restrict__ p, const float* __restrict__ b) {
  int i = blockIdx.x * blockDim.x + threadIdx.x;
  if (i < N_NODES * D) p[i] = b[i & (D - 1)];
}

__device__ __forceinline__ void edge_endpoints(const int* __restrict__ ei, int e,
                                               int& src, int& dst) {
  if (e < E_EDGES) { src = ei[e]; dst = ei[E_EDGES + e]; }
  else             { src = dst = e - E_EDGES; }          // self loop
}

// ---- 4a) alpha_e = leakyrelu(s[src]+d[dst]); per-dst max via atomic ----
__global__ void k_edge_alpha(const int* __restrict__ ei, const float* __restrict__ s,
                             const float* __restrict__ d, float* __restrict__ att,
                             unsigned* __restrict__ m) {
  int e = blockIdx.x * blockDim.x + threadIdx.x;
  if (e >= E_TOT) return;
  int src, dst; edge_endpoints(ei, e, src, dst);
  float a = s[src] + d[dst];
  a = (a > 0.f) ? a : NEG_SLOPE * a;
  att[e] = a;
  atomicMax(m + dst, f2ord(a));
}

// ---- 4b) ex = exp(alpha - max[dst]); denom[dst] += ex (in-place on att) ----
__global__ void k_edge_exp(const int* __restrict__ ei, const unsigned* __restrict__ m,
                           float* __restrict__ att, float* __restrict__ denom) {
  int e = blockIdx.x * blockDim.x + threadIdx.x;
  if (e >= E_TOT) return;
  int src, dst; edge_endpoints(ei, e, src, dst);
  float v = expf(att[e] - ord2f(m[dst]));
  att[e] = v;
  atomicAdd(denom + dst, v);
}

// ---- 4c) out[dst] += (ex/denom[dst]) * h[src]  (one wave per edge) ----
__global__ void k_edge_agg(const int* __restrict__ ei, const float* __restrict__ h,
                           const float* __restrict__ att, const float* __restrict__ denom,
                           float* __restrict__ out) {
  int gw   = (blockIdx.x * blockDim.x + threadIdx.x) >> 5;
  int lane = threadIdx.x & 31;
  if (gw >= E_TOT) return;
  int src, dst; edge_endpoints(ei, gw, src, dst);
  float w = att[gw] / (denom[dst] + 1e-16f);
  const float* hrow = h   + (size_t)src * D;
  float*       orow = out + (size_t)dst * D;
#pragma unroll
  for (int j = 0; j < D / 32; ++j)
    atomicAdd(orow + lane + 32 * j, w * hrow[lane + 32 * j]);
}

// ---- 5) out = gelu(out), exact erf GELU (bias already folded in) ----
__global__ void k_gelu(float* __restrict__ x) {
  int i = blockIdx.x * blockDim.x + threadIdx.x;
  if (i >= N_NODES * D) return;
  float v = x[i];
  x[i] = 0.5f * v * (1.0f + erff(v * 0.70710678118654752f));
}

extern "C" void kernel_launch(void* const* d_in, const int* in_sizes, int n_in,
                              void* d_out, int out_size, void* d_ws, size_t ws_size,
                              hipStream_t stream) {
  const float* x  = (const float*)d_in[0];
  const int*   ei = (const int*)d_in[1];
  // d_in[2]=edge_attr, d_in[3]=batch: unused (GATConv edge_dim=None)
  const float* Wp[3]  = {(const float*)d_in[4],  (const float*)d_in[8],  (const float*)d_in[12]};
  const float* asp[3] = {(const float*)d_in[5],  (const float*)d_in[9],  (const float*)d_in[13]};
  const float* adp[3] = {(const float*)d_in[6],  (const float*)d_in[10], (const float*)d_in[14]};
  const float* bp[3]  = {(const float*)d_in[7],  (const float*)d_in[11], (const float*)d_in[15]};

  const size_t ND = (size_t)N_NODES * D;
  char* ws = (char*)d_ws;
  float*    bufA  = (float*)ws;                  ws += ND * sizeof(float);
  float*    h     = (float*)ws;                  ws += ND * sizeof(float);
  float*    s     = (float*)ws;                  ws += (size_t)N_NODES * sizeof(float);
  float*    dv    = (float*)ws;                  ws += (size_t)N_NODES * sizeof(float);
  unsigned* m     = (unsigned*)ws;               ws += (size_t)N_NODES * sizeof(unsigned);
  float*    denom = (float*)ws;                  ws += (size_t)N_NODES * sizeof(float);
  float*    att   = (float*)ws;                  /* E_TOT floats */
  float*    out   = (float*)d_out;

  // x -> bufA (row-normalized)
  k_normalize<<<N_NODES / 8, 256, 0, stream>>>(x, bufA);

  const float* xin = bufA;
  for (int L = 0; L < 3; ++L) {
    float* xout = (L == 1) ? bufA : out;   // ping-pong: bufA -> out -> bufA -> out

    k_gemm<<<dim3(N_NODES / 16 / MT, 4), dim3(32, 4), 0, stream>>>(xin, Wp[L], h);
    k_sd<<<N_NODES / 8, 256, 0, stream>>>(h, asp[L], adp[L], s, dv);

    k_init_nodes<<<(N_NODES + 255) / 256, 256, 0, stream>>>(m, denom);
    k_init_bias<<<(int)((ND + 255) / 256), 256, 0, stream>>>(xout, bp[L]);

    k_edge_alpha<<<(E_TOT + 255) / 256, 256, 0, stream>>>(ei, s, dv, att, m);
    k_edge_exp  <<<(E_TOT + 255) / 256, 256, 0, stream>>>(ei, m, att, denom);
    k_edge_agg  <<<E_TOT / 8,            256, 0, stream>>>(ei, h, att, denom, xout);

    k_gelu<<<(int)((ND + 255) / 256), 256, 0, stream>>>(xout);
    xin = xout;
  }
}